// GATNet_73469710565760
// MI455X (gfx1250) — compile-verified
//
#include <hip/hip_runtime.h>

typedef __attribute__((ext_vector_type(2))) float v2f;
typedef __attribute__((ext_vector_type(8))) float v8f;

#define F_IN   128
#define HID    32
#define HEADS  4
#define OUTC   16

// ---------------- helpers ----------------
__device__ __forceinline__ unsigned f2ord(float f) {
  // monotonic float -> uint mapping so unsigned atomicMax == float max
  unsigned u = __float_as_uint(f);
  return (u & 0x80000000u) ? ~u : (u | 0x80000000u);
}
__device__ __forceinline__ float ord2f(unsigned u) {
  return (u & 0x80000000u) ? __uint_as_float(u & 0x7fffffffu)
                           : __uint_as_float(~u);
}
__device__ __forceinline__ float lrelu(float v) { return v > 0.f ? v : 0.2f * v; }

// ---------------- zero fill ----------------
__global__ void k_fill0(float* __restrict__ p, long long count) {
  long long i = (long long)blockIdx.x * blockDim.x + threadIdx.x;
  if (i < count) p[i] = 0.f;
}

// ---------------- GEMM1: H[n,128] = x[n,128] @ W[128,128] (f32 WMMA 16x16x4) ----
// block = 256 threads = 8 waves; wave w computes the 16x16 tile at cols [16w,16w+16)
// of the 16-row stripe selected by blockIdx.x.  32 wmma per wave (K=128, step 4).
__global__ void k_gemm1(const float* __restrict__ x, const float* __restrict__ W,
                        float* __restrict__ H, int n) {
  const int wave = threadIdx.x >> 5;
  const int lane = threadIdx.x & 31;
  const int row0 = blockIdx.x * 16;
  if (row0 >= n) return;                 // block-uniform
  const int col0 = wave * 16;
  const int lr = lane & 15;              // M (for A) / N (for B,C)
  const int lh = lane >> 4;              // K half-select
  const float* ap = x + (size_t)(row0 + lr) * F_IN + 2 * lh;
  const float* bp = W + (size_t)(2 * lh) * F_IN + col0 + lr;
  v8f acc = {};
#pragma unroll
  for (int k = 0; k < F_IN; k += 4) {
    // A 16x4: vgpr0 = K=(k+2lh), vgpr1 = K=(k+2lh+1)  (8B aligned -> float2 load)
    v2f a = *(const v2f*)(ap + k);
    // B 4x16: same K split, row-stride F_IN
    v2f b;
    b.x = bp[(size_t)k * F_IN];
    b.y = bp[(size_t)k * F_IN + F_IN];
    acc = __builtin_amdgcn_wmma_f32_16x16x4_f32(false, a, false, b, (short)0, acc,
                                                false, false);
  }
#pragma unroll
  for (int i = 0; i < 8; i++) {
    // C/D layout: vgpr i, lanes 0-15 -> M=i ; lanes 16-31 -> M=i+8
    H[(size_t)(row0 + i + 8 * lh) * F_IN + col0 + lr] = acc[i];
  }
}

// ---------------- GEMM2: H2[n,16] = Hp[n,128] @ W2[128,16] ----------------
// one 16-col tile; each wave owns one 16-row stripe (8 stripes per block).
__global__ void k_gemm2(const float* __restrict__ Hp, const float* __restrict__ W2,
                        float* __restrict__ H2, int n) {
  const int wave = threadIdx.x >> 5;
  const int lane = threadIdx.x & 31;
  const int row0 = (blockIdx.x * 8 + wave) * 16;
  if (row0 >= n) return;                 // wave-uniform
  const int lr = lane & 15;
  const int lh = lane >> 4;
  const float* ap = Hp + (size_t)(row0 + lr) * F_IN + 2 * lh;
  const float* bp = W2 + (size_t)(2 * lh) * OUTC + lr;
  v8f acc = {};
#pragma unroll
  for (int k = 0; k < F_IN; k += 4) {
    v2f a = *(const v2f*)(ap + k);
    v2f b;
    b.x = bp[(size_t)k * OUTC];
    b.y = bp[(size_t)k * OUTC + OUTC];
    acc = __builtin_amdgcn_wmma_f32_16x16x4_f32(false, a, false, b, (short)0, acc,
                                                false, false);
  }
#pragma unroll
  for (int i = 0; i < 8; i++) {
    H2[(size_t)(row0 + i + 8 * lh) * OUTC + lr] = acc[i];
  }
}

// ---------------- per-(node,head) attention coefficient dots ----------------
template <int HN, int CN>
__global__ void k_coef(const float* __restrict__ H, const float* __restrict__ asr,
                       const float* __restrict__ adt, float* __restrict__ als,
                       float* __restrict__ ald, int n) {
  int idx = blockIdx.x * blockDim.x + threadIdx.x;
  if (idx >= n * HN) return;
  int node = idx / HN, h = idx % HN;
  const float* hv = H + (size_t)node * (HN * CN) + h * CN;
  const float* a1 = asr + h * CN;
  const float* a2 = adt + h * CN;
  float s1 = 0.f, s2 = 0.f;
#pragma unroll
  for (int c = 0; c < CN; c++) {
    float v = hv[c];
    s1 += v * a1[c];
    s2 += v * a2[c];
  }
  als[idx] = s1;
  ald[idx] = s2;
}

// ---------------- edge pass A: segment max of leaky-relu logits ----------------
template <int HN>
__global__ void k_edge_max(const int* __restrict__ ei, int E, int n,
                           const float* __restrict__ als, const float* __restrict__ ald,
                           unsigned* __restrict__ smax) {
  int idx = blockIdx.x * blockDim.x + threadIdx.x;
  int tot = (E + n) * HN;
  if (idx >= tot) return;
  int e = idx / HN, h = idx % HN;
  int s, d;
  if (e < E) { s = ei[e]; d = ei[E + e]; } else { s = d = e - E; }
  float logit = lrelu(als[s * HN + h] + ald[d * HN + h]);
  atomicMax(&smax[d * HN + h], f2ord(logit));
}

// ---------------- edge pass B: segment sum of exp(logit - max) ----------------
template <int HN>
__global__ void k_edge_sum(const int* __restrict__ ei, int E, int n,
                           const float* __restrict__ als, const float* __restrict__ ald,
                           const unsigned* __restrict__ smax, float* __restrict__ ssum) {
  int idx = blockIdx.x * blockDim.x + threadIdx.x;
  int tot = (E + n) * HN;
  if (idx >= tot) return;
  int e = idx / HN, h = idx % HN;
  int s, d;
  if (e < E) { s = ei[e]; d = ei[E + e]; } else { s = d = e - E; }
  float logit = lrelu(als[s * HN + h] + ald[d * HN + h]);
  float m = ord2f(smax[d * HN + h]);
  atomicAdd(&ssum[d * HN + h], expf(logit - m));
}

// ---------------- edge pass C: alpha-weighted message scatter-add ----------------
// thread per (edge, head, channel); rem = h*CN + c indexes the head-major feature row
template <int HN, int CN>
__global__ void k_edge_agg(const int* __restrict__ ei, int E, int n,
                           const float* __restrict__ als, const float* __restrict__ ald,
                           const unsigned* __restrict__ smax, const float* __restrict__ ssum,
                           const float* __restrict__ H, float* __restrict__ agg) {
  int idx = blockIdx.x * blockDim.x + threadIdx.x;
  int tot = (E + n) * (HN * CN);
  if (idx >= tot) return;
  int e = idx / (HN * CN);
  int rem = idx - e * (HN * CN);
  int h = rem / CN;
  int s, d;
  if (e < E) { s = ei[e]; d = ei[E + e]; } else { s = d = e - E; }
  float logit = lrelu(als[s * HN + h] + ald[d * HN + h]);
  float m = ord2f(smax[d * HN + h]);
  float alpha = expf(logit - m) / (ssum[d * HN + h] + 1e-16f);
  float msg = H[(size_t)s * (HN * CN) + rem] * alpha;
  atomicAdd(&agg[(size_t)d * (HN * CN) + rem], msg);
}

// ---------------- layer-1 epilogue: bias + ELU + strided column regroup ----------
// new col j = i*32+k  <-  old col 4k+i   (h[:, i::4] concat)
__global__ void k_finish1(const float* __restrict__ agg, const float* __restrict__ b1,
                          float* __restrict__ Hp, int n) {
  int idx = blockIdx.x * blockDim.x + threadIdx.x;
  if (idx >= n * F_IN) return;
  int node = idx >> 7, j = idx & 127;
  int i = j >> 5, k = j & 31;
  int sc = 4 * k + i;
  float v = agg[(size_t)node * F_IN + sc] + b1[sc];
  v = v > 0.f ? v : (expf(v) - 1.f);   // ELU
  Hp[(size_t)node * F_IN + j] = v;
}

// ---------------- layer-2 epilogue: bias + log_softmax (in place on d_out) -------
__global__ void k_finish2(float* __restrict__ out, const float* __restrict__ b2, int n) {
  int node = blockIdx.x * blockDim.x + threadIdx.x;
  if (node >= n) return;
  float v[OUTC];
  float m = -3.0e38f;
#pragma unroll
  for (int c = 0; c < OUTC; c++) {
    v[c] = out[(size_t)node * OUTC + c] + b2[c];
    m = fmaxf(m, v[c]);
  }
  float s = 0.f;
#pragma unroll
  for (int c = 0; c < OUTC; c++) s += expf(v[c] - m);
  float lse = m + logf(s);
#pragma unroll
  for (int c = 0; c < OUTC; c++) out[(size_t)node * OUTC + c] = v[c] - lse;
}

static inline unsigned cdiv(long long a, long long b) { return (unsigned)((a + b - 1) / b); }

extern "C" void kernel_launch(void* const* d_in, const int* in_sizes, int n_in,
                              void* d_out, int out_size, void* d_ws, size_t ws_size,
                              hipStream_t stream) {
  const float* x    = (const float*)d_in[0];
  const int*   ei   = (const int*)  d_in[1];
  const float* W1   = (const float*)d_in[2];
  const float* asr1 = (const float*)d_in[3];
  const float* adt1 = (const float*)d_in[4];
  const float* b1   = (const float*)d_in[5];
  const float* W2   = (const float*)d_in[6];
  const float* asr2 = (const float*)d_in[7];
  const float* adt2 = (const float*)d_in[8];
  const float* b2   = (const float*)d_in[9];
  float* out = (float*)d_out;

  const int n = in_sizes[0] / F_IN;   // 100000
  const int E = in_sizes[1] / 2;      // 1600000

  // workspace layout (floats)
  float* ws = (float*)d_ws;
  size_t off = 0;
  float*    H1    = ws + off; off += (size_t)n * F_IN;   // x@W1, later overwritten by permuted/ELU'd features
  float*    agg1  = ws + off; off += (size_t)n * F_IN;   // layer-1 message accumulator
  float*    als1  = ws + off; off += (size_t)n * HEADS;
  float*    ald1  = ws + off; off += (size_t)n * HEADS;
  unsigned* smax1 = (unsigned*)(ws + off); off += (size_t)n * HEADS;
  float*    ssum1 = ws + off; off += (size_t)n * HEADS;
  float*    H2    = ws + off; off += (size_t)n * OUTC;
  float*    als2  = ws + off; off += (size_t)n;
  float*    ald2  = ws + off; off += (size_t)n;
  unsigned* smax2 = (unsigned*)(ws + off); off += (size_t)n;
  float*    ssum2 = ws + off; off += (size_t)n;

  const int B = 256;
  const long long tot_e1 = (long long)(E + n) * HEADS;
  const long long tot_a1 = (long long)(E + n) * (HEADS * HID);
  const long long tot_e2 = (long long)(E + n);
  const long long tot_a2 = (long long)(E + n) * OUTC;

  // zero accumulators/softmax state (agg1 .. ssum2 are contiguous) + d_out
  long long zcnt = (long long)(agg1 - ws);
  zcnt = (long long)off - zcnt;
  k_fill0<<<cdiv(zcnt, B), B, 0, stream>>>(agg1, zcnt);
  k_fill0<<<cdiv((long long)n * OUTC, B), B, 0, stream>>>(out, (long long)n * OUTC);

  // ---- layer 1 ----
  k_gemm1<<<cdiv(n, 16), B, 0, stream>>>(x, W1, H1, n);
  k_coef<HEADS, HID><<<cdiv((long long)n * HEADS, B), B, 0, stream>>>(H1, asr1, adt1, als1, ald1, n);
  k_edge_max<HEADS><<<cdiv(tot_e1, B), B, 0, stream>>>(ei, E, n, als1, ald1, smax1);
  k_edge_sum<HEADS><<<cdiv(tot_e1, B), B, 0, stream>>>(ei, E, n, als1, ald1, smax1, ssum1);
  k_edge_agg<HEADS, HID><<<cdiv(tot_a1, B), B, 0, stream>>>(ei, E, n, als1, ald1, smax1, ssum1, H1, agg1);
  k_finish1<<<cdiv((long long)n * F_IN, B), B, 0, stream>>>(agg1, b1, H1, n);

  // ---- layer 2 ----
  k_gemm2<<<cdiv(cdiv(n, 16), 8), B, 0, stream>>>(H1, W2, H2, n);
  k_coef<1, OUTC><<<cdiv((long long)n, B), B, 0, stream>>>(H2, asr2, adt2, als2, ald2, n);
  k_edge_max<1><<<cdiv(tot_e2, B), B, 0, stream>>>(ei, E, n, als2, ald2, smax2);
  k_edge_sum<1><<<cdiv(tot_e2, B), B, 0, stream>>>(ei, E, n, als2, ald2, smax2, ssum2);
  k_edge_agg<1, OUTC><<<cdiv(tot_a2, B), B, 0, stream>>>(ei, E, n, als2, ald2, smax2, ssum2, H2, out);
  k_finish2<<<cdiv((long long)n, B), B, 0, stream>>>(out, b2, n);
}